// GCN_encoder_22179211117090
// MI455X (gfx1250) — compile-verified
//
#include <hip/hip_runtime.h>
#include <stdint.h>

typedef __attribute__((ext_vector_type(2))) float v2f;
typedef __attribute__((ext_vector_type(8))) float v8f;

#define N_NODES 50000
#define F_HID   256
#define BN_EPS  1e-5f

// ---------------------------------------------------------------------------
// Pack B[K x 256] (row-major) into per-column-tile interleaved layout:
//   Bp[nblk][k2][l] = { B[2*k2][nblk*16+l], B[2*k2+1][nblk*16+l] }  (float2)
// so each WMMA wave reads one contiguous float2 per lane (coalesced b64).
// ---------------------------------------------------------------------------
__global__ void pack_B(const float* __restrict__ B, float2* __restrict__ Bp, int K)
{
    const int idx = blockIdx.x * blockDim.x + threadIdx.x;   // over 16*(K/2)*16
    const int total = 16 * (K / 2) * 16;
    if (idx >= total) return;
    const int l  = idx & 15;
    const int k2 = (idx >> 4) % (K / 2);
    const int nb = idx / (16 * (K / 2));
    float2 v;
    v.x = B[(size_t)(2 * k2)     * F_HID + nb * 16 + l];
    v.y = B[(size_t)(2 * k2 + 1) * F_HID + nb * 16 + l];
    Bp[idx] = v;
}

// ---------------------------------------------------------------------------
// GEMM  C[M x 256] = A[M x K] @ B[K x 256]  via V_WMMA_F32_16X16X4_F32.
// One block = 16-row tile; 16 waves (512 thr, wave32) = 16 column tiles.
// A tile staged into LDS with global_load_async_to_lds_b128 (ASYNCcnt) --
// each thread issues K/128 16B direct-to-LDS transfers; s_wait_asynccnt + barrier.
// LDS row pad = 4 floats: keeps 16B alignment for b128 and spreads the
// 16-lane row reads across banks (stride 132 dwords == 4 mod 64).
// C may alias A (all A reads land in LDS before any global write; each block
// reads/writes only its own 16 rows).
// ---------------------------------------------------------------------------
template <int K>
__global__ void __launch_bounds__(512)
gemm_wmma_f32(const float* __restrict__ A, const float2* __restrict__ Bp,
              float* __restrict__ C)
{
    constexpr int LDA = K + 4;
    __shared__ float As[16 * LDA];

    const int m0 = blockIdx.x * 16;

    // ---- async stage 16 x K tile of A into LDS -----------------------------
    for (int idx = threadIdx.x; idx < 16 * (K / 4); idx += 512) {
        const int r  = idx / (K / 4);
        const int c4 = (idx % (K / 4)) * 4;
        const unsigned lds_off = (unsigned)(uintptr_t)(&As[r * LDA + c4]);
        const unsigned goff    = (unsigned)(((m0 + r) * K + c4) * 4);
        asm volatile("global_load_async_to_lds_b128 %0, %1, %2"
                     :: "v"(lds_off), "v"(goff), "s"(A)
                     : "memory");
    }
    asm volatile("s_wait_asynccnt 0" ::: "memory");
    __syncthreads();

    const int wave = threadIdx.x >> 5;
    const int lane = threadIdx.x & 31;
    const int half = lane >> 4;                  // 0: {K,K+1}  1: {K+2,K+3}
    const int l    = lane & 15;
    const int n0   = wave * 16;

    const float2* Bw = Bp + (size_t)wave * (K / 2) * 16;

    v8f acc = {};
    for (int k = 0; k < K; k += 4) {
        const int ka = k + 2 * half;             // ka = 2*k2
        v2f a, b;
        a.x = As[l * LDA + ka];
        a.y = As[l * LDA + ka + 1];
        const float2 bv = Bw[(k / 2 + half) * 16 + l];   // coalesced b64
        b.x = bv.x;
        b.y = bv.y;
        acc = __builtin_amdgcn_wmma_f32_16x16x4_f32(
            /*neg_a=*/false, a, /*neg_b=*/false, b,
            /*c_mod=*/(short)0, acc, /*reuse_a=*/false, /*reuse_b=*/false);
    }

#pragma unroll
    for (int r = 0; r < 8; ++r)
        C[(size_t)(m0 + r + 8 * half) * F_HID + n0 + l] = acc[r];
}

// ---------------------------------------------------------------------------
// Degree / normalization
// ---------------------------------------------------------------------------
__global__ void deg_init(float* __restrict__ deg, int n)
{
    const int i = blockIdx.x * blockDim.x + threadIdx.x;
    if (i < n) deg[i] = 1.0f;                    // self loop
}

__global__ void deg_count(const long long* __restrict__ dst,
                          float* __restrict__ deg, int E)
{
    const int e = blockIdx.x * blockDim.x + threadIdx.x;
    if (e < E) atomicAdd(&deg[(int)dst[e]], 1.0f);
}

__global__ void deg_rsqrt(const float* __restrict__ deg,
                          float* __restrict__ dinv, int n)
{
    const int i = blockIdx.x * blockDim.x + threadIdx.x;
    if (i < n) {
        const float d = deg[i];
        dinv[i] = (d > 0.0f) ? rsqrtf(d) : 0.0f;
    }
}

// ---------------------------------------------------------------------------
// Aggregation init: out[i] = h[i]*dinv[i]^2 + bias (self-loop term; also
// overwrites the poisoned destination). float4 vectorized, 4 nodes / block.
// ---------------------------------------------------------------------------
__global__ void __launch_bounds__(256)
agg_init(const float4* __restrict__ h4, const float* __restrict__ dinv,
         const float4* __restrict__ bias4, float4* __restrict__ out4)
{
    const int node = blockIdx.x * 4 + (threadIdx.x >> 6);
    const int c4   = threadIdx.x & 63;           // 64 float4 = 256 cols
    const float s  = dinv[node];
    const float s2 = s * s;
    const float4 hv = h4[(size_t)node * 64 + c4];
    const float4 bv = bias4[c4];
    float4 r;
    r.x = hv.x * s2 + bv.x;
    r.y = hv.y * s2 + bv.y;
    r.z = hv.z * s2 + bv.z;
    r.w = hv.w * s2 + bv.w;
    out4[(size_t)node * 64 + c4] = r;
}

// ---------------------------------------------------------------------------
// Per-edge scatter: one wave per edge (8 edges / 256-thread block, no
// barrier). src/dst scalarized via readfirstlane (SADDR addressing); the
// 256-col row is processed as 8 lane-contiguous 128B slices -> coalesced
// gathers and atomic bursts, all L2-resident (h and out both fit in 192MB L2).
// ---------------------------------------------------------------------------
__global__ void __launch_bounds__(256)
edge_scatter(const long long* __restrict__ src, const long long* __restrict__ dst,
             const float* __restrict__ dinv, const float* __restrict__ h,
             float* __restrict__ out, int E)
{
    const int wave = threadIdx.x >> 5;
    const int lane = threadIdx.x & 31;
    const int e = blockIdx.x * 8 + wave;
    if (e >= E) return;
    const int s = __builtin_amdgcn_readfirstlane((int)src[e]);
    const int d = __builtin_amdgcn_readfirstlane((int)dst[e]);
    const float w = dinv[s] * dinv[d];
    const float* __restrict__ hr = h + (size_t)s * F_HID;
    float* __restrict__ orow = out + (size_t)d * F_HID;
#pragma unroll
    for (int i = 0; i < 8; ++i) {
        const int c = lane + 32 * i;
        atomicAdd(&orow[c], hr[c] * w);          // global_atomic_add_f32
    }
}

// ---------------------------------------------------------------------------
// ReLU + BatchNorm (batch stats, biased variance via E[x^2] - mean^2)
// ---------------------------------------------------------------------------
__global__ void zero_stats(float* __restrict__ stats)
{
    stats[threadIdx.x] = 0.0f;                   // 512 = sum[256] + sumsq[256]
}

__global__ void __launch_bounds__(F_HID)
bn_stats(const float* __restrict__ h, float* __restrict__ stats, int n)
{
    const int c = threadIdx.x;
    float sum = 0.0f, sq = 0.0f;
    for (int r = blockIdx.x; r < n; r += gridDim.x) {
        const int rn = r + gridDim.x;
        if (rn < n) __builtin_prefetch(&h[(size_t)rn * F_HID + c], 0, 0);
        float v = h[(size_t)r * F_HID + c];
        v = fmaxf(v, 0.0f);                      // ReLU before stats
        sum += v;
        sq  += v * v;
    }
    atomicAdd(&stats[c], sum);
    atomicAdd(&stats[F_HID + c], sq);
}

__global__ void __launch_bounds__(F_HID)
bn_apply(const float* __restrict__ h, const float* __restrict__ stats,
         const float* __restrict__ gamma, const float* __restrict__ beta,
         float* __restrict__ out, int n)
{
    const int i = blockIdx.x, c = threadIdx.x;
    const float invN = 1.0f / (float)n;
    const float mean = stats[c] * invN;
    const float var  = stats[F_HID + c] * invN - mean * mean;
    const float v = fmaxf(h[(size_t)i * F_HID + c], 0.0f);
    out[(size_t)i * F_HID + c] =
        (v - mean) * rsqrtf(var + BN_EPS) * gamma[c] + beta[c];
}

// ---------------------------------------------------------------------------
// Orchestration.  d_in: x, edge_index(int64 [2,E]), W1, b1, gamma, beta, W2, b2.
// Workspace: deg | dinv | stats | W1p | W2p | bufA(51.2MB).
// d_out doubles as the layer-1 aggregation buffer; GEMM2 runs in place.
// ---------------------------------------------------------------------------
extern "C" void kernel_launch(void* const* d_in, const int* in_sizes, int n_in,
                              void* d_out, int out_size, void* d_ws, size_t ws_size,
                              hipStream_t stream)
{
    (void)n_in; (void)out_size; (void)ws_size;

    const float*     x     = (const float*)d_in[0];
    const long long* ei    = (const long long*)d_in[1];
    const float*     W1    = (const float*)d_in[2];
    const float*     b1    = (const float*)d_in[3];
    const float*     gamma = (const float*)d_in[4];
    const float*     beta  = (const float*)d_in[5];
    const float*     W2    = (const float*)d_in[6];
    const float*     b2    = (const float*)d_in[7];
    float*           out   = (float*)d_out;

    const int E = in_sizes[1] / 2;
    const long long* src = ei;
    const long long* dst = ei + E;

    char*   w     = (char*)d_ws;
    float*  deg   = (float*)(w + 0);               // 200 KB
    float*  dinv  = (float*)(w + (256 << 10));     // 200 KB
    float*  stats = (float*)(w + (512 << 10));     // 2 KB
    float2* W1p   = (float2*)(w + (640 << 10));    // 128 KB
    float2* W2p   = (float2*)(w + (1 << 20));      // 256 KB
    float*  bufA  = (float*)(w + (2 << 20));       // 51.2 MB

    const int TPB = 256;

    // weight packing (one-time, overlaps nothing else needed)
    pack_B<<<(16 * 64  * 16 + TPB - 1) / TPB, TPB, 0, stream>>>(W1, W1p, 128);
    pack_B<<<(16 * 128 * 16 + TPB - 1) / TPB, TPB, 0, stream>>>(W2, W2p, 256);

    // degrees / D^{-1/2}
    deg_init <<<(N_NODES + TPB - 1) / TPB, TPB, 0, stream>>>(deg, N_NODES);
    deg_count<<<(E + TPB - 1) / TPB, TPB, 0, stream>>>(dst, deg, E);
    deg_rsqrt<<<(N_NODES + TPB - 1) / TPB, TPB, 0, stream>>>(deg, dinv, N_NODES);

    // layer 1: h1 = x @ W1 ; aggregate (+b1) into d_out
    gemm_wmma_f32<128><<<N_NODES / 16, 512, 0, stream>>>(x, W1p, bufA);
    agg_init    <<<N_NODES / 4, TPB, 0, stream>>>((const float4*)bufA, dinv,
                                                  (const float4*)b1, (float4*)out);
    edge_scatter<<<(E + 7) / 8, TPB, 0, stream>>>(src, dst, dinv, bufA, out, E);

    // ReLU + BatchNorm: d_out -> bufA
    zero_stats<<<1, 512, 0, stream>>>(stats);
    bn_stats  <<<1024, F_HID, 0, stream>>>(out, stats, N_NODES);
    bn_apply  <<<N_NODES, F_HID, 0, stream>>>(out, stats, gamma, beta, bufA, N_NODES);

    // layer 2: h2 = hbn @ W2 (in place) ; aggregate (+b2) into d_out
    gemm_wmma_f32<256><<<N_NODES / 16, 512, 0, stream>>>(bufA, W2p, bufA);
    agg_init    <<<N_NODES / 4, TPB, 0, stream>>>((const float4*)bufA, dinv,
                                                  (const float4*)b2, (float4*)out);
    edge_scatter<<<(E + 7) / 8, TPB, 0, stream>>>(src, dst, dinv, bufA, out, E);
}